// AdvancedGCRNN_4105988735443
// MI455X (gfx1250) — compile-verified
//
#include <hip/hip_runtime.h>

// ---------------------------------------------------------------------------
// CDNA5 / gfx1250: GCN x3 -> GAT -> bi-LSTM(step=1) -> BN -> FC.
// Dense matmuls: V_WMMA_F32_16X16X4_F32 (exact fp32, wave32), one 16x16 tile
// per wave, K fully unrolled, ALL B operands pre-transposed so both A and B
// fragments are contiguous global_load_b64 with immediate offsets.
// Edge scatter: L2-resident f32 atomics (GLOBAL_ATOMIC_ADD_F32 /
// GLOBAL_ATOMIC_MAX_U32 via ordered-uint float encoding).
// ---------------------------------------------------------------------------

typedef float v2f __attribute__((ext_vector_type(2)));
typedef float v8f __attribute__((ext_vector_type(8)));

// ---------------- small utility kernels ----------------

__global__ void k_fill_zero(float* __restrict__ p, long long n) {
  long long i = (long long)blockIdx.x * blockDim.x + threadIdx.x;
  long long s = (long long)gridDim.x * blockDim.x;
  for (; i < n; i += s) p[i] = 0.0f;
}

__global__ void k_init_deg(float* __restrict__ deg, int n) {
  int i = blockIdx.x * blockDim.x + threadIdx.x;
  if (i < n) deg[i] = 1.0f;   // self loop
}

__global__ void k_deg_accum(const int* __restrict__ col, int E, float* __restrict__ deg) {
  int i = blockIdx.x * blockDim.x + threadIdx.x;
  if (i < E) atomicAdd(&deg[col[i]], 1.0f);
}

__global__ void k_deg_to_dis(float* __restrict__ deg, int n) {
  int i = blockIdx.x * blockDim.x + threadIdx.x;
  if (i < n) deg[i] = rsqrtf(fmaxf(deg[i], 1e-12f));   // in-place: deg -> dis
}

// W: K x N row-major  ->  WT: N x K row-major  (tiny one-off weight transpose)
template <int LOGN>
__global__ void k_transpose(const float* __restrict__ W, float* __restrict__ WT, int K) {
  constexpr int Nn = 1 << LOGN;
  int idx = blockIdx.x * blockDim.x + threadIdx.x;
  if (idx >= K * Nn) return;
  int k = idx >> LOGN;
  int n = idx & (Nn - 1);
  WT[n * K + k] = W[idx];
}

// ---------------- fp32 WMMA GEMM: C = A(MxK) * BT(NxK)^T ------------------
// One wave owns a 16x16 C tile. Requires M % 16 == 0 (all calls: M = 50000).
// Ragged N (FC: N=10): B row pointer clamped to a valid row; garbage columns
// are never stored. Wave-uniform early exit keeps EXEC all-ones at every
// WMMA (ISA 7.12 requirement). Both fragments load as contiguous v2f with
// compile-time immediate offsets.

template <int K>
__global__ __launch_bounds__(256)
void k_gemm_wmma(const float* __restrict__ A, int lda,
                 const float* __restrict__ BT, int ldb,
                 float* __restrict__ C, int ldc, int M, int N) {
  int wave   = blockIdx.x * (blockDim.x >> 5) + (threadIdx.x >> 5);
  int lane   = threadIdx.x & 31;
  int ntiles = (N + 15) >> 4;
  int mtiles = M >> 4;
  if (wave >= mtiles * ntiles) return;          // wave-uniform

  int tm   = (wave / ntiles) << 4;
  int tn   = (wave % ntiles) << 4;
  int half = lane >> 4;                         // 0 or 1
  int l16  = lane & 15;
  int m    = tm + l16;                          // A row owned by this lane
  int n    = tn + l16;                          // B/C col owned by this lane

  // Lane layouts (ISA 7.12.2): A element (m, k + v + 2*half), v in {0,1};
  // B element (k + v + 2*half, n) = BT[n][k + v + 2*half].
  const float* arow = A  + (long long)m * lda + (half << 1);
  int nc = (n < N) ? n : (N - 1);               // clamp keeps loads in-bounds
  const float* brow = BT + (long long)nc * ldb + (half << 1);

  v8f acc = {0.f, 0.f, 0.f, 0.f, 0.f, 0.f, 0.f, 0.f};
#pragma unroll
  for (int k = 0; k < K; k += 4) {
    v2f a = *(const v2f*)(arow + k);            // 8B aligned (even kk, even lda)
    v2f b = *(const v2f*)(brow + k);
    acc = __builtin_amdgcn_wmma_f32_16x16x4_f32(false, a, false, b,
                                                (short)0, acc, false, false);
  }

  bool nv = (n < N);
#pragma unroll
  for (int v = 0; v < 8; ++v) {                 // C row = v + 8*half (ISA 7.12.2)
    int r = tm + v + (half << 3);
    if (nv) C[(long long)r * ldc + n] = acc[v];
  }
}

// ---------------- GCN edge aggregation + epilogue (F compile-time) ---------

template <int LOGF4>
__global__ void k_gcn_aggregate(const int* __restrict__ row, const int* __restrict__ col, int E,
                                const float* __restrict__ tmp, const float* __restrict__ dis,
                                float* __restrict__ accum) {
  constexpr int F4 = 1 << LOGF4;
  long long total = (long long)E << LOGF4;
  long long s = (long long)gridDim.x * blockDim.x;
  for (long long i = (long long)blockIdx.x * blockDim.x + threadIdx.x; i < total; i += s) {
    int e = (int)(i >> LOGF4);
    int q = (int)(i & (F4 - 1));
    int r = row[e], c = col[e];
    float w = dis[r] * dis[c];
    float4 v = ((const float4*)tmp)[((long long)r << LOGF4) + q];
    float* dst = accum + (((long long)c << LOGF4) + q) * 4;
    atomicAdd(dst + 0, v.x * w);
    atomicAdd(dst + 1, v.y * w);
    atomicAdd(dst + 2, v.z * w);
    atomicAdd(dst + 3, v.w * w);
  }
}

template <int LOGF>
__global__ void k_gcn_epilogue(const float* __restrict__ accum, const float* __restrict__ tmp,
                               const float* __restrict__ dis, const float* __restrict__ bias,
                               float* __restrict__ out, int N, int relu) {
  constexpr int F = 1 << LOGF;
  long long total = (long long)N << LOGF;
  long long s = (long long)gridDim.x * blockDim.x;
  for (long long i = (long long)blockIdx.x * blockDim.x + threadIdx.x; i < total; i += s) {
    int node = (int)(i >> LOGF);
    int f    = (int)(i & (F - 1));
    float d  = dis[node];
    float v  = accum[i] + tmp[i] * (d * d) + bias[f];   // self-loop norm = dis^2
    out[i] = relu ? fmaxf(v, 0.0f) : v;
  }
}

// ---------------- GAT ----------------

__device__ __forceinline__ unsigned f2ord(float f) {
  unsigned u = __float_as_uint(f);
  return (u & 0x80000000u) ? ~u : (u | 0x80000000u);
}
__device__ __forceinline__ float ord2f(unsigned u) {
  return (u & 0x80000000u) ? __uint_as_float(u & 0x7FFFFFFFu) : __uint_as_float(~u);
}
__device__ __forceinline__ float lrelu(float x) { return x > 0.0f ? x : 0.2f * x; }

__global__ void k_gat_scores(const float* __restrict__ g, const float* __restrict__ a_src,
                             const float* __restrict__ a_dst, float* __restrict__ s_src,
                             float* __restrict__ s_dst, unsigned* __restrict__ m_ord, int N) {
  int i = blockIdx.x * blockDim.x + threadIdx.x;
  if (i >= N) return;
  const float* gi = g + (long long)i * 32;
  float ss = 0.0f, sd = 0.0f;
#pragma unroll
  for (int f = 0; f < 32; ++f) { float v = gi[f]; ss += v * a_src[f]; sd += v * a_dst[f]; }
  s_src[i] = ss;
  s_dst[i] = sd;
  m_ord[i] = f2ord(lrelu(ss + sd));             // self-edge seeds the segment max
}

__global__ void k_gat_edge_max(const int* __restrict__ row, const int* __restrict__ col, int E,
                               const float* __restrict__ s_src, const float* __restrict__ s_dst,
                               unsigned* __restrict__ m_ord) {
  int e = blockIdx.x * blockDim.x + threadIdx.x;
  if (e >= E) return;
  int r = row[e], c = col[e];
  atomicMax(&m_ord[c], f2ord(lrelu(s_src[r] + s_dst[c])));
}

__global__ void k_gat_node_init(const float* __restrict__ g, const float* __restrict__ s_src,
                                const float* __restrict__ s_dst, const unsigned* __restrict__ m_ord,
                                float* __restrict__ m, float* __restrict__ ssum,
                                float* __restrict__ num, int N) {
  int i = blockIdx.x * blockDim.x + threadIdx.x;
  if (i >= N) return;
  float mm = ord2f(m_ord[i]);
  m[i] = mm;
  float w = __expf(lrelu(s_src[i] + s_dst[i]) - mm);   // self-edge weight
  ssum[i] = w;
  const float* gi = g + (long long)i * 32;
  float* ni = num + (long long)i * 32;
#pragma unroll
  for (int f = 0; f < 32; ++f) ni[f] = gi[f] * w;      // self-edge numerator
}

__global__ void k_gat_edge_sum(const int* __restrict__ row, const int* __restrict__ col, int E,
                               const float* __restrict__ s_src, const float* __restrict__ s_dst,
                               const float* __restrict__ m, float* __restrict__ ee,
                               float* __restrict__ ssum) {
  int e = blockIdx.x * blockDim.x + threadIdx.x;
  if (e >= E) return;
  int r = row[e], c = col[e];
  float w = __expf(lrelu(s_src[r] + s_dst[c]) - m[c]);
  ee[e] = w;
  atomicAdd(&ssum[c], w);
}

__global__ void k_gat_edge_num(const int* __restrict__ row, const int* __restrict__ col, int E,
                               const float* __restrict__ g, const float* __restrict__ ee,
                               float* __restrict__ num) {
  long long total = (long long)E * 8;                  // 8 float4 chunks of 32 feats
  long long s = (long long)gridDim.x * blockDim.x;
  for (long long i = (long long)blockIdx.x * blockDim.x + threadIdx.x; i < total; i += s) {
    int e = (int)(i >> 3);
    int q = (int)(i & 7);
    int r = row[e], c = col[e];
    float w = ee[e];
    float4 v = ((const float4*)g)[(long long)r * 8 + q];
    float* dst = num + (long long)c * 32 + q * 4;
    atomicAdd(dst + 0, v.x * w);
    atomicAdd(dst + 1, v.y * w);
    atomicAdd(dst + 2, v.z * w);
    atomicAdd(dst + 3, v.w * w);
  }
}

__global__ void k_gat_epilogue(const float* __restrict__ num, const float* __restrict__ ssum,
                               const float* __restrict__ bias, float* __restrict__ out, int N) {
  int i = blockIdx.x * blockDim.x + threadIdx.x;
  if (i >= N * 32) return;
  int node = i >> 5, f = i & 31;
  out[i] = fmaxf(num[i] / ssum[node] + bias[f], 0.0f);
}

// ---------------- LSTM cell (seq_len=1, h0=c0=0) + BatchNorm fold ----------

__global__ void k_lstm_elem(const float* __restrict__ gt, const float* __restrict__ bih,
                            const float* __restrict__ bhh, const float* __restrict__ gamma,
                            const float* __restrict__ beta, float* __restrict__ r,
                            int N, int chOff) {
  int idx = blockIdx.x * blockDim.x + threadIdx.x;
  if (idx >= N * 64) return;
  int node = idx >> 6, j = idx & 63;
  const float* gr = gt + (long long)node * 256;
  float ig = gr[j]       + bih[j]       + bhh[j];        // gate order i,f,g,o; f dead (c0=0)
  float cg = gr[128 + j] + bih[128 + j] + bhh[128 + j];
  float og = gr[192 + j] + bih[192 + j] + bhh[192 + j];
  float si = 1.0f / (1.0f + __expf(-ig));
  float so = 1.0f / (1.0f + __expf(-og));
  float val = so * tanhf(si * tanhf(cg));
  int ch = chOff + j;
  float scale = gamma[ch] * rsqrtf(1.0f + 1e-5f);        // BN: mean 0, var 1
  r[(long long)node * 128 + ch] = val * scale + beta[ch];
}

__global__ void k_bias_out(float* __restrict__ out, const float* __restrict__ b, int total) {
  int i = blockIdx.x * blockDim.x + threadIdx.x;
  if (i < total) out[i] += b[i % 10];                    // NCLS = 10 (compile-time)
}

// ---------------------------------------------------------------------------

extern "C" void kernel_launch(void* const* d_in, const int* in_sizes, int n_in,
                              void* d_out, int out_size, void* d_ws, size_t ws_size,
                              hipStream_t stream) {
  (void)n_in; (void)out_size; (void)ws_size;

  const float* x     = (const float*)d_in[0];
  const int*   ei    = (const int*)d_in[1];
  const float* W1    = (const float*)d_in[2];
  const float* b1    = (const float*)d_in[3];
  const float* W2    = (const float*)d_in[4];
  const float* b2    = (const float*)d_in[5];
  const float* W3    = (const float*)d_in[6];
  const float* b3    = (const float*)d_in[7];
  const float* gatW  = (const float*)d_in[8];
  const float* aSrc  = (const float*)d_in[9];
  const float* aDst  = (const float*)d_in[10];
  const float* gatB  = (const float*)d_in[11];
  const float* WihF  = (const float*)d_in[12];
  const float* bihF  = (const float*)d_in[13];
  const float* bhhF  = (const float*)d_in[14];
  const float* WihB  = (const float*)d_in[15];
  const float* bihB  = (const float*)d_in[16];
  const float* bhhB  = (const float*)d_in[17];
  const float* gamma = (const float*)d_in[18];
  const float* beta  = (const float*)d_in[19];
  const float* fcW   = (const float*)d_in[20];
  const float* fcB   = (const float*)d_in[21];

  const int N = in_sizes[0] / 128;     // 50000  (multiple of 16)
  const int E = in_sizes[1] / 2;       // 800000
  const int* row = ei;                 // edge_index[0] = source
  const int* col = ei + E;             // edge_index[1] = target

  // ---- workspace bump allocator (~107 MB total) ----
  char* wp = (char*)d_ws;
  auto alloc = [&](size_t bytes) -> void* {
    void* r = wp; wp += (bytes + 255) & ~(size_t)255; return r;
  };
  float*    BIG   = (float*)alloc((size_t)N * 256 * 4);  // GEMM scratch / LSTM gates
  float*    HB0   = (float*)alloc((size_t)N * 128 * 4);  // h features
  float*    HB1   = (float*)alloc((size_t)N * 128 * 4);  // accumulators / r
  float*    dis   = (float*)alloc((size_t)N * 4);
  float*    sSrc  = (float*)alloc((size_t)N * 4);
  float*    sDst  = (float*)alloc((size_t)N * 4);
  float*    mArr  = (float*)alloc((size_t)N * 4);
  float*    ssum  = (float*)alloc((size_t)N * 4);
  unsigned* mOrd  = (unsigned*)alloc((size_t)N * 4);
  float*    eeArr = (float*)alloc((size_t)E * 4);
  float*    W1T   = (float*)alloc((size_t)128 * 128 * 4);
  float*    W2T   = (float*)alloc((size_t)128 * 64 * 4);
  float*    W3T   = (float*)alloc((size_t)64 * 32 * 4);
  float*    gatWT = (float*)alloc((size_t)32 * 32 * 4);

  const int BLK = 256;
  auto blocks = [&](long long t) { return (int)((t + BLK - 1) / BLK); };
  auto gemm_grid = [&](int M, int Nn) {
    return (((M + 15) / 16) * ((Nn + 15) / 16) + 7) / 8;
  };

  // ---- one-off weight transposes (W is KxN -> WT is NxK) ----
  k_transpose<7><<<blocks(128 * 128), BLK, 0, stream>>>(W1, W1T, 128);
  k_transpose<6><<<blocks(128 * 64),  BLK, 0, stream>>>(W2, W2T, 128);
  k_transpose<5><<<blocks(64 * 32),   BLK, 0, stream>>>(W3, W3T, 64);
  k_transpose<5><<<blocks(32 * 32),   BLK, 0, stream>>>(gatW, gatWT, 32);

  // ---- degrees and symmetric normalization ----
  k_init_deg<<<blocks(N), BLK, 0, stream>>>(dis, N);
  k_deg_accum<<<blocks(E), BLK, 0, stream>>>(col, E, dis);
  k_deg_to_dis<<<blocks(N), BLK, 0, stream>>>(dis, N);

  // ---- GCN layer 1: 128 -> 128, relu ----
  k_gemm_wmma<128><<<gemm_grid(N, 128), 256, 0, stream>>>(x, 128, W1T, 128, BIG, 128, N, 128);
  k_fill_zero<<<blocks((long long)N * 128), BLK, 0, stream>>>(HB1, (long long)N * 128);
  k_gcn_aggregate<5><<<blocks((long long)E * 32), BLK, 0, stream>>>(row, col, E, BIG, dis, HB1);
  k_gcn_epilogue<7><<<blocks((long long)N * 128), BLK, 0, stream>>>(HB1, BIG, dis, b1, HB0, N, 1);

  // ---- GCN layer 2: 128 -> 64, relu ----
  k_gemm_wmma<128><<<gemm_grid(N, 64), 256, 0, stream>>>(HB0, 128, W2T, 128, BIG, 64, N, 64);
  k_fill_zero<<<blocks((long long)N * 64), BLK, 0, stream>>>(HB1, (long long)N * 64);
  k_gcn_aggregate<4><<<blocks((long long)E * 16), BLK, 0, stream>>>(row, col, E, BIG, dis, HB1);
  k_gcn_epilogue<6><<<blocks((long long)N * 64), BLK, 0, stream>>>(HB1, BIG, dis, b2, HB0, N, 1);

  // ---- GCN layer 3: 64 -> 32, no relu ----
  k_gemm_wmma<64><<<gemm_grid(N, 32), 256, 0, stream>>>(HB0, 64, W3T, 64, BIG, 32, N, 32);
  k_fill_zero<<<blocks((long long)N * 32), BLK, 0, stream>>>(HB1, (long long)N * 32);
  k_gcn_aggregate<3><<<blocks((long long)E * 8), BLK, 0, stream>>>(row, col, E, BIG, dis, HB1);
  k_gcn_epilogue<5><<<blocks((long long)N * 32), BLK, 0, stream>>>(HB1, BIG, dis, b3, HB0, N, 0);

  // ---- GAT (1 head, softmax over incoming edges incl. self loop) ----
  k_gemm_wmma<32><<<gemm_grid(N, 32), 256, 0, stream>>>(HB0, 32, gatWT, 32, BIG, 32, N, 32);
  k_gat_scores<<<blocks(N), BLK, 0, stream>>>(BIG, aSrc, aDst, sSrc, sDst, mOrd, N);
  k_gat_edge_max<<<blocks(E), BLK, 0, stream>>>(row, col, E, sSrc, sDst, mOrd);
  k_gat_node_init<<<blocks(N), BLK, 0, stream>>>(BIG, sSrc, sDst, mOrd, mArr, ssum, HB1, N);
  k_gat_edge_sum<<<blocks(E), BLK, 0, stream>>>(row, col, E, sSrc, sDst, mArr, eeArr, ssum);
  k_gat_edge_num<<<blocks((long long)E * 8), BLK, 0, stream>>>(row, col, E, BIG, eeArr, HB1);
  k_gat_epilogue<<<blocks((long long)N * 32), BLK, 0, stream>>>(HB1, ssum, gatB, HB0, N);

  // ---- bidirectional LSTM (single step) + folded BatchNorm -> r in HB1 ----
  // lstm_Wih is stored (256,32) row-major == B^T layout already.
  k_gemm_wmma<32><<<gemm_grid(N, 256), 256, 0, stream>>>(HB0, 32, WihF, 32, BIG, 256, N, 256);
  k_lstm_elem<<<blocks((long long)N * 64), BLK, 0, stream>>>(BIG, bihF, bhhF, gamma, beta, HB1, N, 0);
  k_gemm_wmma<32><<<gemm_grid(N, 256), 256, 0, stream>>>(HB0, 32, WihB, 32, BIG, 256, N, 256);
  k_lstm_elem<<<blocks((long long)N * 64), BLK, 0, stream>>>(BIG, bihB, bhhB, gamma, beta, HB1, N, 64);

  // ---- final FC: out = r @ fc_W^T + fc_b  (fc_W is (10,128) == B^T) ----
  k_gemm_wmma<128><<<gemm_grid(N, 10), 256, 0, stream>>>(HB1, 128, fcW, 128, (float*)d_out, 10, N, 10);
  k_bias_out<<<blocks((long long)N * 10), BLK, 0, stream>>>((float*)d_out, fcB, N * 10);
}